// DotProductAttention_9010841387791
// MI455X (gfx1250) — compile-verified
//
#include <hip/hip_runtime.h>
#include <math.h>

typedef float v2f __attribute__((ext_vector_type(2)));
typedef float v8f __attribute__((ext_vector_type(8)));

#define B_ 16
#define Q_ 2048
#define K_ 2048
#define D_ 512

// ---- kernel 1 tiling: 512 thr = 16 waves, 32 q-rows x 2048 keys per block ----
#define QT1 32         // q-rows per block (2 row-tiles of 16)
#define RT1 2          // row tiles per wave
#define NW1 16         // waves per block
#define CT1 8          // 16-col key tiles per wave (8*16 = 128 cols/wave)
#define KPW1 (CT1*16)  // key cols per wave
#define QPAD 516       // padded LDS row stride (floats): %64==4 -> conflict-free b64 frags

// ---- kernel 2 tiling: 256 thr = 8 waves, 64 q-rows x 512 d-cols per block ----
#define QT2 64         // q-rows per block (4 row-tiles of 16)
#define RT2 4          // row tiles per wave
#define NW2 8          // waves per block
#define CT2 4          // 16-col d tiles per wave (4*16 = 64 cols/wave)
#define KCHUNK 128     // K staging chunk
#define WPAD 132       // padded LDS row stride for weights chunk

__device__ __forceinline__ v8f wmma_f32(v2f a, v2f b, v8f c) {
  // D = A(16x4 f32) * B(4x16 f32) + C(16x16 f32)
  return __builtin_amdgcn_wmma_f32_16x16x4_f32(false, a, false, b, (short)0, c,
                                               false, false);
}

// ---------------------------------------------------------------------------
// Kernel 1: scores = Q*K^T * scale, mask, softmax -> weights [B,Q,K]
// Each K B-fragment is reused across RT1 row-tiles (halves L2 B-traffic).
// ---------------------------------------------------------------------------
__global__ __launch_bounds__(512)
void attn_scores_softmax(const float* __restrict__ qin,
                         const float* __restrict__ kin,
                         const int* __restrict__ valid_lens,
                         float* __restrict__ weights) {
  __shared__ float qtile[QT1 * QPAD];          // 32 x 512 staged Q rows
  __shared__ float redmax[NW1 * QT1];
  __shared__ float redsum[NW1 * QT1];

  const int b    = blockIdx.y;
  const int q0   = blockIdx.x * QT1;
  const int tid  = threadIdx.x;
  const int wave = tid >> 5;
  const int lane = tid & 31;
  const int n    = lane & 15;   // N (and M-within-tile) index
  const int hi   = lane >> 4;   // half select
  const int vlen = valid_lens[b];

  // ---- stage Q tile (32 x 512 f32) into LDS, padded rows ----
  {
    const float4* qg = (const float4*)(qin + ((size_t)b * Q_ + q0) * D_);
    for (int i = tid; i < QT1 * (D_ / 4); i += 512) {
      int row = i >> 7;          // 128 float4 per row
      int c4  = i & 127;
      float4 val = qg[row * (D_ / 4) + c4];
      float* dst = &qtile[row * QPAD + c4 * 4];
      dst[0] = val.x; dst[1] = val.y; dst[2] = val.z; dst[3] = val.w;
    }
  }
  __syncthreads();

  const int    kbase = wave * KPW1;
  const float* kb    = kin + (size_t)b * K_ * D_;

  v8f acc[RT1][CT1] = {};

  // ---- GEMM over D in chunks of 32 (8 k-steps of 4) ----
  for (int dc = 0; dc < D_; dc += 32) {
    v2f afrag[RT1][8];
#pragma unroll
    for (int rt = 0; rt < RT1; ++rt)
#pragma unroll
      for (int i = 0; i < 8; ++i) {
        int d = dc + i * 4 + hi * 2;
        afrag[rt][i] = *(const v2f*)&qtile[(rt * 16 + n) * QPAD + d];
      }
#pragma unroll
    for (int t = 0; t < CT1; ++t) {
      const float* kcol = kb + (size_t)(kbase + t * 16 + n) * D_;
#pragma unroll
      for (int i = 0; i < 8; ++i) {
        int d = dc + i * 4 + hi * 2;
        v2f bfrag = *(const v2f*)&kcol[d];           // loaded once...
#pragma unroll
        for (int rt = 0; rt < RT1; ++rt)             // ...used RT1 times
          acc[rt][t] = wmma_f32(afrag[rt][i], bfrag, acc[rt][t]);
      }
    }
  }

  // ---- scale + length mask ----
  const float scale = 0.04419417382415922f;  // 1/sqrt(512)
#pragma unroll
  for (int t = 0; t < CT1; ++t) {
    int col = kbase + t * 16 + n;
    bool masked = col >= vlen;
#pragma unroll
    for (int rt = 0; rt < RT1; ++rt)
#pragma unroll
      for (int j = 0; j < 8; ++j) {
        float s = acc[rt][t][j] * scale;
        acc[rt][t][j] = masked ? -__builtin_inff() : s;
      }
  }

  // ---- row max within wave, then across waves via LDS ----
  float rm[RT1][8];
#pragma unroll
  for (int rt = 0; rt < RT1; ++rt)
#pragma unroll
    for (int j = 0; j < 8; ++j) rm[rt][j] = -__builtin_inff();
#pragma unroll
  for (int t = 0; t < CT1; ++t)
#pragma unroll
    for (int rt = 0; rt < RT1; ++rt)
#pragma unroll
      for (int j = 0; j < 8; ++j) rm[rt][j] = fmaxf(rm[rt][j], acc[rt][t][j]);
#pragma unroll
  for (int rt = 0; rt < RT1; ++rt)
#pragma unroll
    for (int j = 0; j < 8; ++j)
#pragma unroll
      for (int m = 1; m <= 8; m <<= 1)
        rm[rt][j] = fmaxf(rm[rt][j], __shfl_xor(rm[rt][j], m, 32));
  if (n == 0) {  // lane 0: rows rt*16+0..7, lane 16: rows rt*16+8..15
#pragma unroll
    for (int rt = 0; rt < RT1; ++rt)
#pragma unroll
      for (int j = 0; j < 8; ++j)
        redmax[wave * QT1 + rt * 16 + hi * 8 + j] = rm[rt][j];
  }
  __syncthreads();

  float gm[RT1][8];
#pragma unroll
  for (int rt = 0; rt < RT1; ++rt)
#pragma unroll
    for (int j = 0; j < 8; ++j) {
      float m = -__builtin_inff();
#pragma unroll
      for (int w = 0; w < NW1; ++w)
        m = fmaxf(m, redmax[w * QT1 + rt * 16 + hi * 8 + j]);
      gm[rt][j] = m;
    }

  // ---- exponentiate + row sums ----
  float rs[RT1][8];
#pragma unroll
  for (int rt = 0; rt < RT1; ++rt)
#pragma unroll
    for (int j = 0; j < 8; ++j) rs[rt][j] = 0.f;
#pragma unroll
  for (int t = 0; t < CT1; ++t)
#pragma unroll
    for (int rt = 0; rt < RT1; ++rt)
#pragma unroll
      for (int j = 0; j < 8; ++j) {
        float p = __expf(acc[rt][t][j] - gm[rt][j]);  // exp(-inf-x)=0 -> mask
        acc[rt][t][j] = p;
        rs[rt][j] += p;
      }
#pragma unroll
  for (int rt = 0; rt < RT1; ++rt)
#pragma unroll
    for (int j = 0; j < 8; ++j)
#pragma unroll
      for (int m = 1; m <= 8; m <<= 1)
        rs[rt][j] += __shfl_xor(rs[rt][j], m, 32);
  if (n == 0) {
#pragma unroll
    for (int rt = 0; rt < RT1; ++rt)
#pragma unroll
      for (int j = 0; j < 8; ++j)
        redsum[wave * QT1 + rt * 16 + hi * 8 + j] = rs[rt][j];
  }
  __syncthreads();

  float inv[RT1][8];
#pragma unroll
  for (int rt = 0; rt < RT1; ++rt)
#pragma unroll
    for (int j = 0; j < 8; ++j) {
      float s = 0.f;
#pragma unroll
      for (int w = 0; w < NW1; ++w)
        s += redsum[w * QT1 + rt * 16 + hi * 8 + j];
      inv[rt][j] = 1.0f / s;   // >= 1: max element contributes exp(0)=1
    }

  // ---- write normalized weights stripes ----
  float* wout = weights + ((size_t)b * Q_ + q0) * (size_t)K_;
#pragma unroll
  for (int t = 0; t < CT1; ++t) {
    int col = kbase + t * 16 + n;
#pragma unroll
    for (int rt = 0; rt < RT1; ++rt)
#pragma unroll
      for (int j = 0; j < 8; ++j) {
        int row = rt * 16 + hi * 8 + j;
        wout[(size_t)row * K_ + col] = acc[rt][t][j] * inv[rt][j];
      }
  }
}

// ---------------------------------------------------------------------------
// Kernel 2: context = weights * V -> [B,Q,D]
// 64 q-rows x 512 d-cols per block; each V B-fragment reused across RT2=4
// row-tiles (quarters L2 V-traffic).
// ---------------------------------------------------------------------------
__global__ __launch_bounds__(256)
void attn_context(const float* __restrict__ weights,
                  const float* __restrict__ vin,
                  float* __restrict__ out) {
  __shared__ float wtile[QT2 * WPAD];   // 64 x 128 staged weights chunk

  const int b    = blockIdx.y;
  const int q0   = blockIdx.x * QT2;
  const int tid  = threadIdx.x;
  const int wave = tid >> 5;
  const int lane = tid & 31;
  const int n    = lane & 15;
  const int hi   = lane >> 4;

  const float* wrow  = weights + ((size_t)b * Q_ + q0) * (size_t)K_;
  const float* vbase = vin + (size_t)b * K_ * D_;

  v8f acc[RT2][CT2] = {};

  for (int kc = 0; kc < K_; kc += KCHUNK) {
    // stage 64 x 128 weights chunk into LDS
    for (int i = tid; i < QT2 * (KCHUNK / 4); i += 256) {
      int row = i >> 5;          // 32 float4 per row
      int c4  = i & 31;
      const float4* src = (const float4*)&wrow[(size_t)row * K_ + kc];
      float4 val = src[c4];
      float* dst = &wtile[row * WPAD + c4 * 4];
      dst[0] = val.x; dst[1] = val.y; dst[2] = val.z; dst[3] = val.w;
    }
    __syncthreads();

#pragma unroll 2
    for (int i = 0; i < KCHUNK / 4; ++i) {
      v2f afrag[RT2];
#pragma unroll
      for (int rt = 0; rt < RT2; ++rt)
        afrag[rt] = *(const v2f*)&wtile[(rt * 16 + n) * WPAD + i * 4 + hi * 2];
#pragma unroll
      for (int t = 0; t < CT2; ++t) {
        int dcol = wave * (CT2 * 16) + t * 16 + n;
        int kk   = kc + i * 4 + hi * 2;
        const float* vp = vbase + (size_t)kk * D_ + dcol;
        v2f bfrag;
        bfrag.x = vp[0];     // V[kk][dcol]
        bfrag.y = vp[D_];    // V[kk+1][dcol]
#pragma unroll
        for (int rt = 0; rt < RT2; ++rt)   // one V fragment -> 4 WMMAs
          acc[rt][t] = wmma_f32(afrag[rt], bfrag, acc[rt][t]);
      }
    }
    __syncthreads();
  }

  float* orow = out + ((size_t)b * Q_ + q0) * (size_t)D_;
#pragma unroll
  for (int t = 0; t < CT2; ++t) {
    int dcol = wave * (CT2 * 16) + t * 16 + n;
#pragma unroll
    for (int rt = 0; rt < RT2; ++rt)
#pragma unroll
      for (int j = 0; j < 8; ++j)
        orow[(size_t)(rt * 16 + hi * 8 + j) * D_ + dcol] = acc[rt][t][j];
  }
}

// ---------------------------------------------------------------------------
extern "C" void kernel_launch(void* const* d_in, const int* in_sizes, int n_in,
                              void* d_out, int out_size, void* d_ws, size_t ws_size,
                              hipStream_t stream) {
  const float* q  = (const float*)d_in[0];
  const float* k  = (const float*)d_in[1];
  const float* v  = (const float*)d_in[2];
  const int*   vl = (const int*)d_in[3];

  float* ctx = (float*)d_out;                       // context first in tuple
  float* wts = ctx + (size_t)B_ * Q_ * D_;          // weights second

  dim3 grid1(Q_ / QT1, B_);
  attn_scores_softmax<<<grid1, 512, 0, stream>>>(q, k, vl, wts);

  dim3 grid2(Q_ / QT2, B_);
  attn_context<<<grid2, 256, 0, stream>>>(wts, v, ctx);
}